// GATEncoder_69483980914698
// MI455X (gfx1250) — compile-verified
//
#include <hip/hip_runtime.h>
#include <math.h>

typedef __attribute__((ext_vector_type(2))) float v2f;
typedef __attribute__((ext_vector_type(8))) float v8f;

// ---------- helpers: order-preserving float <-> uint key for atomicMax-based segment_max ----------
__device__ __forceinline__ unsigned f2key(float f) {
    unsigned b = __float_as_uint(f);
    return (b & 0x80000000u) ? ~b : (b | 0x80000000u);
}
__device__ __forceinline__ float key2f(unsigned k) {
    unsigned b = (k & 0x80000000u) ? (k & 0x7FFFFFFFu) : ~k;
    return __uint_as_float(b);
}

// ---------- WMMA f32 GEMM: C[M,NC] = A[M,K] * B[K,NC], row-major. K%4==0, NC%16==0. ----------
// One wave32 owns 4 consecutive... rather: block = 4 waves, wave w handles M-tile (blockIdx*4+w)
// across ALL N-tiles. B is staged transposed in LDS (Bs[n*K+k]) so each lane's B-fragment is a
// contiguous float2 -> ds_load_b64. A-fragments are contiguous float2 -> global_load_b64, cached
// in registers (K/4 v2f) and reused across N-tiles. Row guard by clamping (no EXEC predication:
// WMMA requires EXEC all-ones; clamped rows compute garbage that is never stored).
//
// V_WMMA_F32_16X16X4_F32 layout (ISA 7.12.2):
//   A 16x4: lanes 0-15 -> M=lane, VGPRs = K0,K1 ; lanes 16-31 -> M=lane-16, VGPRs = K2,K3
//   B 4x16 / C,D 16x16: row striped across lanes within a VGPR; lane halves offset.
template <int K, int NC>
__global__ void gemm_wmma_f32(const float* __restrict__ A,
                              const float* __restrict__ B,
                              float* __restrict__ C,
                              int M) {
    __shared__ float Bs[NC * K];   // transposed weights: Bs[n*K + k] = B[k*NC + n]
    const int tid = threadIdx.x;

    // cooperative transposed stage of B (coalesced global reads)
    for (int idx = tid; idx < K * NC; idx += 128) {
        int k = idx / NC;
        int n = idx - k * NC;
        Bs[n * K + k] = B[idx];
    }
    __syncthreads();

    const int waveId = tid >> 5;
    const int lane   = tid & 31;
    const int half   = lane >> 4;       // 0 | 1
    const int l15    = lane & 15;
    const int kb     = half << 1;       // K offset 0 | 2
    const int m0     = (blockIdx.x * 4 + waveId) << 4;
    if (m0 >= M) return;                // wave-uniform exit (after barrier)

    int rowA = m0 + l15;
    if (rowA >= M) rowA = M - 1;        // clamp: padded rows computed but never stored
    const float* __restrict__ arow = A + (size_t)rowA * K;

    v2f areg[K / 4];
#pragma unroll
    for (int kk = 0; kk < K / 4; ++kk)
        areg[kk] = *(const v2f*)(arow + kk * 4 + kb);

#pragma unroll
    for (int tn = 0; tn < NC / 16; ++tn) {
        const int n0 = tn << 4;
        const float* bcol = &Bs[(n0 + l15) * K + kb];
        v8f acc = {};
#pragma unroll
        for (int kk = 0; kk < K / 4; ++kk) {
            v2f b = *(const v2f*)(bcol + kk * 4);
            acc = __builtin_amdgcn_wmma_f32_16x16x4_f32(false, areg[kk], false, b,
                                                        (short)0, acc, false, false);
        }
        const int colC  = n0 + l15;
        const int rbase = m0 + (half << 3);
#pragma unroll
        for (int r = 0; r < 8; ++r) {
            int row = rbase + r;
            if (row < M) C[(size_t)row * NC + colC] = acc[r];
        }
    }
}

// ---------- per-(node,head) attention scores: as[n,h] = <h[n,h,:], att_src[h,:]>, same for ad ----------
__global__ void att_scores(const float* __restrict__ h,
                           const float* __restrict__ a_src,
                           const float* __restrict__ a_dst,
                           float* __restrict__ out_s,
                           float* __restrict__ out_d,
                           int N, int H, int Cc) {
    int t = blockIdx.x * blockDim.x + threadIdx.x;
    if (t >= N * H) return;
    int n  = t / H;
    int hh = t - n * H;
    const float* row = h + (size_t)n * H * Cc + (size_t)hh * Cc;
    const float* vs  = a_src + (size_t)hh * Cc;
    const float* vd  = a_dst + (size_t)hh * Cc;
    float ss = 0.f, dd = 0.f;
    for (int c = 0; c < Cc; ++c) {
        float v = row[c];
        ss += v * vs[c];
        dd += v * vd[c];
    }
    out_s[t] = ss;
    out_d[t] = dd;
}

// ---------- edge pass A: leaky-relu logit, cache it, segment_max via uint-key atomicMax ----------
__global__ void edge_logit_max(const int* __restrict__ ei, int E, int H,
                               const float* __restrict__ as_, const float* __restrict__ ad_,
                               float* __restrict__ p, unsigned* __restrict__ mkey) {
    int e = blockIdx.x * blockDim.x + threadIdx.x;
    if (e >= E) return;
    int s = ei[e];
    int d = ei[E + e];
    for (int h = 0; h < H; ++h) {
        float el = as_[s * H + h] + ad_[d * H + h];
        el = (el > 0.f) ? el : 0.2f * el;          // LeakyReLU(0.2)
        p[(size_t)e * H + h] = el;
        atomicMax(&mkey[d * H + h], f2key(el));
    }
}

// ---------- edge pass B: p = exp(e - m[dst]); segment_sum via atomicAdd ----------
__global__ void edge_softmax_num(const int* __restrict__ ei, int E, int H,
                                 float* __restrict__ p,
                                 const unsigned* __restrict__ mkey,
                                 float* __restrict__ ssum) {
    int e = blockIdx.x * blockDim.x + threadIdx.x;
    if (e >= E) return;
    int d = ei[E + e];
    for (int h = 0; h < H; ++h) {
        float el = p[(size_t)e * H + h];
        float m  = key2f(mkey[d * H + h]);
        float pe = __expf(el - m);
        p[(size_t)e * H + h] = pe;
        atomicAdd(&ssum[d * H + h], pe);
    }
}

// ---------- edge pass C: out[dst] += h[src] * alpha ; one thread per (edge, channel) ----------
// logF = log2(H*C), logC = log2(C); consecutive threads hit consecutive channels (coalesced).
__global__ void edge_aggregate(const int* __restrict__ ei, int E,
                               int logF, int logC,
                               const float* __restrict__ p,
                               const float* __restrict__ ssum,
                               const float* __restrict__ hsrc,
                               float* __restrict__ out) {
    long long t = (long long)blockIdx.x * blockDim.x + threadIdx.x;
    long long total = (long long)E << logF;
    if (t >= total) return;
    int e  = (int)(t >> logF);
    int c  = (int)(t & ((1 << logF) - 1));
    int hh = c >> logC;
    int H  = 1 << (logF - logC);
    int s  = ei[e];
    int d  = ei[E + e];
    float alpha = p[(size_t)e * H + hh] / (ssum[(size_t)d * H + hh] + 1e-16f);
    atomicAdd(&out[((size_t)d << logF) + c], hsrc[((size_t)s << logF) + c] * alpha);
}

// ---------- x = elu(x + bias) ----------
__global__ void bias_elu(float* __restrict__ x, const float* __restrict__ b,
                         long long total, int mask) {
    long long t = (long long)blockIdx.x * blockDim.x + threadIdx.x;
    if (t >= total) return;
    float v = x[t] + b[(int)(t & mask)];
    x[t] = (v > 0.f) ? v : (__expf(v) - 1.f);
}

// ---------- out += bias ----------
__global__ void bias_add(float* __restrict__ x, const float* __restrict__ b,
                         long long total, int mask) {
    long long t = (long long)blockIdx.x * blockDim.x + threadIdx.x;
    if (t >= total) return;
    x[t] += b[(int)(t & mask)];
}

extern "C" void kernel_launch(void* const* d_in, const int* in_sizes, int n_in,
                              void* d_out, int out_size, void* d_ws, size_t ws_size,
                              hipStream_t stream) {
    const float* x        = (const float*)d_in[0];
    const int*   ei       = (const int*)  d_in[1];
    const float* W1       = (const float*)d_in[2];
    const float* att_src1 = (const float*)d_in[3];
    const float* att_dst1 = (const float*)d_in[4];
    const float* b1       = (const float*)d_in[5];
    const float* W2       = (const float*)d_in[6];
    const float* att_src2 = (const float*)d_in[7];
    const float* att_dst2 = (const float*)d_in[8];
    const float* b2       = (const float*)d_in[9];
    float* out = (float*)d_out;

    const int N = in_sizes[0] / 128;   // 50000 nodes
    const int E = in_sizes[1] / 2;     // 800000 edges

    // ---- workspace layout (floats). h2 aliases h1, p2 aliases p1 (dead by then). ----
    float* ws = (float*)d_ws;
    float*    h1  = ws;                                  // [N,128] = x@W1
    float*    f1  = h1 + (size_t)N * 128;                // [N,128] layer-1 agg, then elu'd feats
    float*    as1 = f1 + (size_t)N * 128;                // [N,4]
    float*    ad1 = as1 + (size_t)N * 4;                 // [N,4]
    unsigned* m1  = (unsigned*)(ad1 + (size_t)N * 4);    // [N,4] max keys
    float*    s1  = (float*)m1 + (size_t)N * 4;          // [N,4]
    float*    as2 = s1 + (size_t)N * 4;                  // [N]
    float*    ad2 = as2 + N;                             // [N]
    unsigned* m2  = (unsigned*)(ad2 + N);                // [N]
    float*    s2  = (float*)m2 + N;                      // [N]
    float*    p1  = s2 + N;                              // [E,4]
    float*    h2  = h1;                                  // [N,64] alias (h1 dead after agg1)
    float*    p2  = p1;                                  // [E]    alias (p1 dead after agg1)

    // zero accumulators: f1 + all small buffers are one contiguous block; key 0 == "-inf" for m*
    hipMemsetAsync(f1, 0, sizeof(float) * ((size_t)N * 128 + (size_t)N * 20), stream);
    hipMemsetAsync(out, 0, sizeof(float) * (size_t)N * 64, stream);

    const int TB = 256;
    const int mTiles = (N + 15) / 16;
    const int gBlocks = (mTiles + 3) / 4;   // 4 waves (4 M-tiles) per block

    // ======== Layer 1: heads=4, C=32, F=128 ========
    gemm_wmma_f32<128, 128><<<gBlocks, 128, 0, stream>>>(x, W1, h1, N);
    att_scores<<<(N * 4 + TB - 1) / TB, TB, 0, stream>>>(h1, att_src1, att_dst1, as1, ad1, N, 4, 32);
    edge_logit_max  <<<(E + TB - 1) / TB, TB, 0, stream>>>(ei, E, 4, as1, ad1, p1, m1);
    edge_softmax_num<<<(E + TB - 1) / TB, TB, 0, stream>>>(ei, E, 4, p1, m1, s1);
    {
        long long total = (long long)E << 7;             // E * 128
        edge_aggregate<<<(int)((total + TB - 1) / TB), TB, 0, stream>>>(ei, E, 7, 5, p1, s1, h1, f1);
    }
    {
        long long total = (long long)N * 128;
        bias_elu<<<(int)((total + TB - 1) / TB), TB, 0, stream>>>(f1, b1, total, 127);
    }

    // ======== Layer 2: heads=1, C=64, F=64 ========
    gemm_wmma_f32<128, 64><<<gBlocks, 128, 0, stream>>>(f1, W2, h2, N);
    att_scores<<<(N + TB - 1) / TB, TB, 0, stream>>>(h2, att_src2, att_dst2, as2, ad2, N, 1, 64);
    edge_logit_max  <<<(E + TB - 1) / TB, TB, 0, stream>>>(ei, E, 1, as2, ad2, p2, m2);
    edge_softmax_num<<<(E + TB - 1) / TB, TB, 0, stream>>>(ei, E, 1, p2, m2, s2);
    {
        long long total = (long long)E << 6;             // E * 64
        edge_aggregate<<<(int)((total + TB - 1) / TB), TB, 0, stream>>>(ei, E, 6, 6, p2, s2, h2, out);
    }
    {
        long long total = (long long)N * 64;             // heads=1 -> mean is identity; just add bias
        bias_add<<<(int)((total + TB - 1) / TB), TB, 0, stream>>>(out, b2, total, 63);
    }
}